// GraphAutoencoder_91302414778554
// MI455X (gfx1250) — compile-verified
//
#include <hip/hip_runtime.h>

// ---------------------------------------------------------------------------
// GAT autoencoder for MI455X (gfx1250, wave32).
//   Per layer:  W-transpose -> GEMM (f32 WMMA 16x16x4) -> att dots ->
//               edge max -> edge exp+sum -> scatter-aggregate -> bias+ReLU
// Bound by the 850K-edge / 128-channel gather+atomic-scatter (~0.9 GB of L2
// traffic per layer; whole working set lives in the 192MB L2). GEMM is
// ~1.6 GFLOP/layer, so full-precision f32 WMMA is free.
// ---------------------------------------------------------------------------

#define NNODES   50000
#define NEDGES   800000
#define ETOTAL   (NNODES + NEDGES)
#define FDIM     128
#define NEGSLOPE 0.2f

typedef __attribute__((ext_vector_type(2))) float v2f;
typedef __attribute__((ext_vector_type(8))) float v8f;

#if defined(__gfx1250__) && __has_builtin(__builtin_amdgcn_wmma_f32_16x16x4_f32)
#define HAVE_WMMA_F32 1
#endif

// ---------------------------------------------------------------------------
// WT[n,k] = W[k,n]  (64KB, once per layer; makes the WMMA B-fragment a
// contiguous 8-byte load identical in shape to the A-fragment).
// ---------------------------------------------------------------------------
__global__ void transpose128_k(const float* __restrict__ W,
                               float* __restrict__ WT) {
  int gid = blockIdx.x * blockDim.x + threadIdx.x;   // 16384 threads
  int k = gid >> 7, n = gid & 127;
  WT[(size_t)n * FDIM + k] = W[(size_t)k * FDIM + n];
}

// ---------------------------------------------------------------------------
// D[N,128] = A[N,128] @ W[128,128] with W given transposed (WT[n,k]).
// One wave -> one 16x16 tile via V_WMMA_F32_16X16X4_F32 (K-loop of 32).
// Block = 8 waves = one 16-row stripe x all 8 col tiles (A rows L0-reused).
// A-frag (ISA 7.12.2, 32-bit A 16x4): lane&15 = M, lane>>4 selects K-pair
// {0,1} vs {2,3}. B-frag mirrors it on WT rows. C/D: VGPR r ->
// M = r + 8*(lane>>4), N = lane&15.
// ---------------------------------------------------------------------------
__global__ __launch_bounds__(256) void gemm128_wmma(const float* __restrict__ A,
                                                    const float* __restrict__ WT,
                                                    float* __restrict__ D) {
  const int lane  = threadIdx.x & 31;
  const int wave  = threadIdx.x >> 5;
  const int row0  = blockIdx.x << 4;      // N = 50000 = 3125 * 16, exact
  const int col0  = wave << 4;
  const int m     = lane & 15;
  const int hi    = lane >> 4;
  const int rbase = hi << 3;

  const float* __restrict__ brow = WT + (size_t)(col0 + m) * FDIM;

#ifdef HAVE_WMMA_F32
  const int koff = hi << 1;               // 0 or 2
  const float* __restrict__ arow = A + (size_t)(row0 + m) * FDIM;
  v8f acc = {};
  for (int k0 = 0; k0 < FDIM; k0 += 4) {
    v2f a = *(const v2f*)(arow + k0 + koff);   // A[m][k .. k+1]
    v2f b = *(const v2f*)(brow + k0 + koff);   // W[k .. k+1][n] via WT
    acc = __builtin_amdgcn_wmma_f32_16x16x4_f32(false, a, false, b,
                                                (short)0, acc, false, false);
  }
#pragma unroll
  for (int r = 0; r < 8; ++r)
    D[(size_t)(row0 + rbase + r) * FDIM + col0 + m] = acc[r];
#else
  for (int r = 0; r < 8; ++r) {
    const float* __restrict__ ar = A + (size_t)(row0 + rbase + r) * FDIM;
    float s = 0.f;
    for (int k = 0; k < FDIM; ++k) s += ar[k] * brow[k];
    D[(size_t)(row0 + rbase + r) * FDIM + col0 + m] = s;
  }
#endif
}

// ---------------------------------------------------------------------------
// a_src[n,h] = <H[n,h,:], att_src[h,:]>, a_dst likewise.  heads*C == 128.
// float4 loads (C is a multiple of 4).
// ---------------------------------------------------------------------------
__global__ void att_dots_k(const float* __restrict__ H,
                           const float* __restrict__ att_src,
                           const float* __restrict__ att_dst,
                           float* __restrict__ a_src,
                           float* __restrict__ a_dst,
                           int heads, int C, int total) {
  int gid = blockIdx.x * blockDim.x + threadIdx.x;
  if (gid >= total) return;
  int n = gid / heads, hd = gid - n * heads;
  const float4* __restrict__ hp  =
      (const float4*)(H + (size_t)n * FDIM + hd * C);
  const float4* __restrict__ asp = (const float4*)(att_src + hd * C);
  const float4* __restrict__ adp = (const float4*)(att_dst + hd * C);
  float s = 0.f, d = 0.f;
  for (int c = 0; c < (C >> 2); ++c) {
    float4 h = hp[c], va = asp[c], vd = adp[c];
    s += h.x * va.x + h.y * va.y + h.z * va.z + h.w * va.w;
    d += h.x * vd.x + h.y * vd.y + h.z * vd.z + h.w * vd.w;
  }
  a_src[gid] = s;
  a_dst[gid] = d;
}

// ---- edge helpers ---------------------------------------------------------
__device__ __forceinline__ void edge_sd(const long long* __restrict__ ei,
                                        int e, int& s, int& d) {
  if (e < NEDGES) { s = (int)ei[e]; d = (int)ei[NEDGES + e]; }
  else            { s = d = e - NEDGES; }            // appended self-loops
}
__device__ __forceinline__ float lrelu(float x) {
  return x > 0.f ? x : NEGSLOPE * x;
}
// monotone float<->u32 mapping so float-max == unsigned-max
__device__ __forceinline__ unsigned fkey(float f) {
  unsigned u = __float_as_uint(f);
  return (u & 0x80000000u) ? ~u : (u | 0x80000000u);
}
__device__ __forceinline__ float funkey(unsigned u) {
  return __uint_as_float((u & 0x80000000u) ? (u & 0x7FFFFFFFu) : ~u);
}

// ---------------------------------------------------------------------------
// Pass 1: segment max of leaky_relu(a_src[src]+a_dst[dst]) per (dst, head).
// emax pre-zeroed (0 is below every real key; self-loops guarantee >=1 hit).
// ---------------------------------------------------------------------------
__global__ void edge_max_k(const long long* __restrict__ ei,
                           const float* __restrict__ a_src,
                           const float* __restrict__ a_dst,
                           unsigned* __restrict__ emax,
                           int heads, int total) {
  int gid = blockIdx.x * blockDim.x + threadIdx.x;
  if (gid >= total) return;
  int e = gid / heads, hd = gid - e * heads;
  int s, d; edge_sd(ei, e, s, d);
  float logit = lrelu(a_src[s * heads + hd] + a_dst[d * heads + hd]);
  atomicMax(&emax[d * heads + hd], fkey(logit));
}

// ---------------------------------------------------------------------------
// Pass 2: e_exp = exp(logit - max), denom[dst,h] += e_exp  (denom pre-zeroed)
// ---------------------------------------------------------------------------
__global__ void edge_expsum_k(const long long* __restrict__ ei,
                              const float* __restrict__ a_src,
                              const float* __restrict__ a_dst,
                              const unsigned* __restrict__ emax,
                              float* __restrict__ e_exp,
                              float* __restrict__ denom,
                              int heads, int total) {
  int gid = blockIdx.x * blockDim.x + threadIdx.x;
  if (gid >= total) return;
  int e = gid / heads, hd = gid - e * heads;
  int s, d; edge_sd(ei, e, s, d);
  float logit = lrelu(a_src[s * heads + hd] + a_dst[d * heads + hd]);
  float v = __expf(logit - funkey(emax[d * heads + hd]));
  e_exp[gid] = v;
  atomicAdd(&denom[d * heads + hd], v);
}

// ---------------------------------------------------------------------------
// Pass 3: acc[dst, c] += H[src, c] * e_exp[e,h] / (denom[dst,h] + 1e-16)
// One thread per (edge, 4-channel quad): one b128 gather + 4 f32 atomics.
// A 4-aligned quad never crosses a head boundary (heads split at mult. of 32),
// so e_exp/denom are loaded once per thread. hd = c0 >> cshift.
// ---------------------------------------------------------------------------
__global__ void aggregate_k(const long long* __restrict__ ei,
                            const float* __restrict__ H,
                            const float* __restrict__ e_exp,
                            const float* __restrict__ denom,
                            float* __restrict__ acc,
                            int heads, int cshift, long long total) {
  long long gid = (long long)blockIdx.x * blockDim.x + threadIdx.x;
  if (gid >= total) return;
  int e  = (int)(gid >> 5);
  int c0 = (int)(gid & 31) << 2;
  int hd = c0 >> cshift;
  int s, d; edge_sd(ei, e, s, d);
  const float* __restrict__ hp = H + (size_t)s * FDIM + c0;
  __builtin_prefetch(hp, 0, 3);          // global_prefetch_b8: overlap gather
  float alpha = e_exp[(size_t)e * heads + hd] /
                (denom[d * heads + hd] + 1e-16f);
  float4 h = *(const float4*)hp;
  float* __restrict__ ap = acc + (size_t)d * FDIM + c0;
  atomicAdd(ap + 0, h.x * alpha);
  atomicAdd(ap + 1, h.y * alpha);
  atomicAdd(ap + 2, h.z * alpha);
  atomicAdd(ap + 3, h.w * alpha);
}

// ---------------------------------------------------------------------------
// out = relu(acc + bias), float4-wide.
// ---------------------------------------------------------------------------
__global__ void bias_relu_k(const float* __restrict__ acc,
                            const float* __restrict__ bias,
                            float* __restrict__ out, int total4) {
  int gid = blockIdx.x * blockDim.x + threadIdx.x;
  if (gid >= total4) return;
  int c0 = (gid & 31) << 2;
  float4 v = ((const float4*)acc)[gid];
  float4 b = *(const float4*)(bias + c0);
  v.x = fmaxf(v.x + b.x, 0.f);
  v.y = fmaxf(v.y + b.y, 0.f);
  v.z = fmaxf(v.z + b.z, 0.f);
  v.w = fmaxf(v.w + b.w, 0.f);
  ((float4*)out)[gid] = v;
}

// ---------------------------------------------------------------------------
extern "C" void kernel_launch(void* const* d_in, const int* in_sizes, int n_in,
                              void* d_out, int out_size, void* d_ws, size_t ws_size,
                              hipStream_t stream) {
  const float*     x   = (const float*)d_in[0];
  const long long* ei  = (const long long*)d_in[1];   // int64 [2, E]
  const float*     W1  = (const float*)d_in[2];
  const float*     as1 = (const float*)d_in[3];
  const float*     ad1 = (const float*)d_in[4];
  const float*     b1  = (const float*)d_in[5];
  const float*     W2  = (const float*)d_in[6];
  const float*     as2 = (const float*)d_in[7];
  const float*     ad2 = (const float*)d_in[8];
  const float*     b2  = (const float*)d_in[9];
  float*           out = (float*)d_out;

  // workspace layout (floats)
  float*    ws    = (float*)d_ws;
  float*    Hbuf  = ws;                                // N*128   (25.6 MB)
  float*    Zbuf  = Hbuf + (size_t)NNODES * FDIM;      // N*128
  float*    Acc   = Zbuf + (size_t)NNODES * FDIM;      // N*128
  float*    aS    = Acc  + (size_t)NNODES * FDIM;      // N*4
  float*    aD    = aS   + (size_t)NNODES * 4;         // N*4
  unsigned* eMax  = (unsigned*)(aD + (size_t)NNODES * 4); // N*4
  float*    Den   = (float*)eMax + (size_t)NNODES * 4; // N*4
  float*    Eexp  = Den  + (size_t)NNODES * 4;         // ETOTAL*4 (13.6 MB)
  float*    WT    = Eexp + (size_t)ETOTAL * 4;         // 128*128  (64 KB)

  const int       B         = 256;
  const int       gemm_grid = NNODES / 16;                 // 3125
  const long long aggN      = (long long)ETOTAL * 32;      // quads: 27.2M
  const int       agg_grid  = (int)((aggN + B - 1) / B);
  const int       nq        = NNODES * 32;                 // output quads
  const int       nq_grid   = (nq + B - 1) / B;

  auto layer = [&](const float* X, const float* W, const float* as,
                   const float* ad, const float* bias, float* dst,
                   int heads, int C, int cshift) {
    const int nh = NNODES * heads;
    const int eh = ETOTAL * heads;
    hipMemsetAsync(eMax, 0, (size_t)nh * sizeof(unsigned), stream);
    hipMemsetAsync(Den,  0, (size_t)nh * sizeof(float), stream);
    hipMemsetAsync(Acc,  0, (size_t)NNODES * FDIM * sizeof(float), stream);

    transpose128_k<<<(FDIM * FDIM) / B, B, 0, stream>>>(W, WT);
    gemm128_wmma<<<gemm_grid, B, 0, stream>>>(X, WT, Hbuf);
    att_dots_k<<<(nh + B - 1) / B, B, 0, stream>>>(Hbuf, as, ad, aS, aD,
                                                   heads, C, nh);
    edge_max_k<<<(eh + B - 1) / B, B, 0, stream>>>(ei, aS, aD, eMax, heads, eh);
    edge_expsum_k<<<(eh + B - 1) / B, B, 0, stream>>>(ei, aS, aD, eMax, Eexp,
                                                      Den, heads, eh);
    aggregate_k<<<agg_grid, B, 0, stream>>>(ei, Hbuf, Eexp, Den, Acc,
                                            heads, cshift, aggN);
    bias_relu_k<<<nq_grid, B, 0, stream>>>(Acc, bias, dst, nq);
  };

  // encoder: heads=4, C=32 ; decoder: heads=1, C=128
  layer(x,    W1, as1, ad1, b1, Zbuf, 4, 32,  5);
  layer(Zbuf, W2, as2, ad2, b2, out,  1, 128, 7);
}